// WeightedRigidAlign_50337016709383
// MI455X (gfx1250) — compile-verified
//
#include <hip/hip_runtime.h>
#include <stdint.h>

typedef __attribute__((ext_vector_type(2))) float v2f;
typedef __attribute__((ext_vector_type(8))) float v8f;
typedef __attribute__((ext_vector_type(4))) uint32_t u32x4;
typedef __attribute__((ext_vector_type(8))) uint32_t u32x8;

#define NPTS 512
#define TPB  256

__device__ __forceinline__ void jacobi3(float G[3][3], float V[3][3], int p, int q, int r) {
  float apq = G[p][q];
  if (apq * apq > 1e-36f) {
    float app = G[p][p], aqq = G[q][q];
    float theta = (aqq - app) * 0.5f * __builtin_amdgcn_rcpf(apq);
    float t = copysignf(1.0f, theta) / (fabsf(theta) + sqrtf(1.0f + theta * theta));
    float c = rsqrtf(1.0f + t * t);
    float s = t * c;
    float apr = G[p][r], aqr = G[q][r];
    G[p][p] = app - t * apq;
    G[q][q] = aqq + t * apq;
    G[p][q] = 0.0f; G[q][p] = 0.0f;
    float nrp = c * apr - s * aqr;
    float nrq = s * apr + c * aqr;
    G[p][r] = nrp; G[r][p] = nrp;
    G[q][r] = nrq; G[r][q] = nrq;
#pragma unroll
    for (int i = 0; i < 3; ++i) {
      float vip = V[i][p], viq = V[i][q];
      V[i][p] = c * vip - s * viq;
      V[i][q] = s * vip + c * viq;
    }
  }
}

// det-preserving column swap (swap + negate = 90-degree rotation)
__device__ __forceinline__ void colswap(float V[3][3], int i, int j) {
#pragma unroll
  for (int r = 0; r < 3; ++r) {
    float t = V[r][i];
    V[r][i] = V[r][j];
    V[r][j] = -t;
  }
}

__global__ __launch_bounds__(TPB) void wra_kernel(const float* __restrict__ pred,
                                                  const float* __restrict__ tru,
                                                  const float* __restrict__ wgt,
                                                  const unsigned char* __restrict__ msk,
                                                  float* __restrict__ out) {
  __shared__ float sT[NPTS * 3];   // true coords staged by TDM, [n][3]
  __shared__ float acc[16];        // 9 cov products, 3 w*t, 3 w*p, wsum

  const int tid = threadIdx.x;
  const int bb  = blockIdx.x;

  if (tid < 16) acc[tid] = 0.0f;

  // ---- TDM: one wave DMAs this batch's true_coords (1536 f32 = 6 KB) straight into LDS ----
  if (tid < 32) {
    const uint64_t gaddr = (uint64_t)(uintptr_t)(tru + (size_t)bb * (NPTS * 3));
    const uint32_t laddr = (uint32_t)(size_t)(void*)&sT[0];
    u32x4 g0;
    g0[0] = 1u;                                                     // count=1, no gather
    g0[1] = laddr;                                                  // lds_addr (bytes)
    g0[2] = (uint32_t)gaddr;                                        // global_addr[31:0]
    g0[3] = ((uint32_t)(gaddr >> 32) & 0x01FFFFFFu) | (2u << 30);   // addr[56:32] | type=2
    u32x8 g1;
    g1[0] = 0x00020000u;                    // workgroup_mask=0, data_size=4B
    g1[1] = (uint32_t)(NPTS * 3) << 16;     // tensor_dim0 = 1536
    g1[2] = 1u << 16;                       // tensor_dim1 = 1
    g1[3] = (uint32_t)(NPTS * 3) << 16;     // tile_dim0 = 1536
    g1[4] = 1u;                             // tile_dim1 = 1, tile_dim2 = 0
    g1[5] = (uint32_t)(NPTS * 3);           // tensor_dim0_stride[31:0] = 1536
    g1[6] = (uint32_t)(NPTS * 3) << 16;     // tensor_dim1_stride[15:0]
    g1[7] = 0u;
    asm volatile("tensor_load_to_lds %0, %1" : : "s"(g0), "s"(g1) : "memory");
  }

  // ---- overlap with the DMA: stream pred / weights / mask (read-once -> non-temporal) ----
  const size_t base2 = (size_t)bb * (NPTS * 3 / 2);
  const v2f* p2 = (const v2f*)pred + base2 + 3 * tid;
  v2f pA = __builtin_nontemporal_load(p2 + 0);
  v2f pB = __builtin_nontemporal_load(p2 + 1);
  v2f pC = __builtin_nontemporal_load(p2 + 2);
  v2f w2 = __builtin_nontemporal_load((const v2f*)wgt + (size_t)bb * (NPTS / 2) + tid);
  const unsigned char* mp = msk + (size_t)bb * NPTS + 2 * tid;
  const float m0 = mp[0] ? 1.0f : 0.0f;
  const float m1 = mp[1] ? 1.0f : 0.0f;

  __builtin_amdgcn_s_wait_tensorcnt(0);   // no-op for waves that issued nothing
  __syncthreads();                        // TDM data + acc zeroing visible to all waves

  // ---- read this thread's 2 true points from LDS, apply mask ----
  const v2f* sT2 = (const v2f*)sT + 3 * tid;
  v2f lA = sT2[0], lB = sT2[1], lC = sT2[2];
  const float t0x = lA.x * m0, t0y = lA.y * m0, t0z = lB.x * m0;
  const float t1x = lB.y * m1, t1y = lC.x * m1, t1z = lC.y * m1;
  const float p0x = pA.x, p0y = pA.y, p0z = pB.x;
  const float p1x = pB.y, p1y = pC.x, p1z = pC.y;
  const float w0 = w2.x * m0, w1 = w2.y * m1;

  // masked-out points must read as 0 in the WMMA apply phase (no-op when mask is all ones)
  float* st = sT + 6 * tid;
  if (m0 == 0.0f) { st[0] = 0.0f; st[1] = 0.0f; st[2] = 0.0f; }
  if (m1 == 0.0f) { st[3] = 0.0f; st[4] = 0.0f; st[5] = 0.0f; }

  // ---- per-thread partial sums (shifted covariance formula) ----
  const float a0x = w0 * t0x, a0y = w0 * t0y, a0z = w0 * t0z;
  const float a1x = w1 * t1x, a1y = w1 * t1y, a1z = w1 * t1z;

  float r16[16];
  r16[0] = a0x * p0x + a1x * p1x;
  r16[1] = a0x * p0y + a1x * p1y;
  r16[2] = a0x * p0z + a1x * p1z;
  r16[3] = a0y * p0x + a1y * p1x;
  r16[4] = a0y * p0y + a1y * p1y;
  r16[5] = a0y * p0z + a1y * p1z;
  r16[6] = a0z * p0x + a1z * p1x;
  r16[7] = a0z * p0y + a1z * p1y;
  r16[8] = a0z * p0z + a1z * p1z;
  r16[9]  = a0x + a1x;
  r16[10] = a0y + a1y;
  r16[11] = a0z + a1z;
  r16[12] = w0 * p0x + w1 * p1x;
  r16[13] = w0 * p0y + w1 * p1y;
  r16[14] = w0 * p0z + w1 * p1z;
  r16[15] = w0 + w1;

  // ---- wave32 butterfly reduction, then cross-wave LDS float atomics ----
#pragma unroll
  for (int i = 0; i < 16; ++i) {
    float v = r16[i];
    v += __shfl_xor(v, 1, 32);
    v += __shfl_xor(v, 2, 32);
    v += __shfl_xor(v, 4, 32);
    v += __shfl_xor(v, 8, 32);
    v += __shfl_xor(v, 16, 32);
    r16[i] = v;
  }
  if ((tid & 31) == 0) {
#pragma unroll
    for (int i = 0; i < 16; ++i) atomicAdd(&acc[i], r16[i]);
  }
  __syncthreads();

  // ---- every thread redundantly (wave-uniform) builds cov and its 3x3 Kabsch rotation ----
  const float inv = 1.0f / acc[15];
  const float tb0 = acc[9] * inv, tb1 = acc[10] * inv, tb2 = acc[11] * inv;
  const float qx = acc[12] * inv, qy = acc[13] * inv, qz = acc[14] * inv;
  float C[3][3] = {
    { acc[0] - acc[9]  * qx, acc[1] - acc[9]  * qy, acc[2] - acc[9]  * qz },
    { acc[3] - acc[10] * qx, acc[4] - acc[10] * qy, acc[5] - acc[10] * qz },
    { acc[6] - acc[11] * qx, acc[7] - acc[11] * qy, acc[8] - acc[11] * qz } };

  float G[3][3];
#pragma unroll
  for (int i = 0; i < 3; ++i)
#pragma unroll
    for (int j = 0; j < 3; ++j)
      G[i][j] = C[0][i] * C[0][j] + C[1][i] * C[1][j] + C[2][i] * C[2][j];

  float V[3][3] = { {1.f,0.f,0.f}, {0.f,1.f,0.f}, {0.f,0.f,1.f} };
#pragma unroll
  for (int sw = 0; sw < 4; ++sw) {
    jacobi3(G, V, 0, 1, 2);
    jacobi3(G, V, 0, 2, 1);
    jacobi3(G, V, 1, 2, 0);
  }

  float d0 = G[0][0], d1 = G[1][1], d2 = G[2][2];
  if (d0 < d1) { float td = d0; d0 = d1; d1 = td; colswap(V, 0, 1); }
  if (d0 < d2) { float td = d0; d0 = d2; d2 = td; colswap(V, 0, 2); }
  if (d1 < d2) { float td = d1; d1 = d2; d2 = td; colswap(V, 1, 2); }

  float u0[3], u1[3], u2[3];
#pragma unroll
  for (int i = 0; i < 3; ++i)
    u0[i] = C[i][0] * V[0][0] + C[i][1] * V[1][0] + C[i][2] * V[2][0];
  float n0 = u0[0]*u0[0] + u0[1]*u0[1] + u0[2]*u0[2];
  float in0 = rsqrtf(fmaxf(n0, 1e-30f));
  u0[0] *= in0; u0[1] *= in0; u0[2] *= in0;
#pragma unroll
  for (int i = 0; i < 3; ++i)
    u1[i] = C[i][0] * V[0][1] + C[i][1] * V[1][1] + C[i][2] * V[2][1];
  float dp = u1[0]*u0[0] + u1[1]*u0[1] + u1[2]*u0[2];
  u1[0] -= dp * u0[0]; u1[1] -= dp * u0[1]; u1[2] -= dp * u0[2];
  float n1 = u1[0]*u1[0] + u1[1]*u1[1] + u1[2]*u1[2];
  float in1 = rsqrtf(fmaxf(n1, 1e-30f));
  u1[0] *= in1; u1[1] *= in1; u1[2] *= in1;
  u2[0] = u0[1]*u1[2] - u0[2]*u1[1];
  u2[1] = u0[2]*u1[0] - u0[0]*u1[2];
  u2[2] = u0[0]*u1[1] - u0[1]*u1[0];   // right-handed U => reflection fix is automatic

  float R[3][3];
#pragma unroll
  for (int i = 0; i < 3; ++i)
#pragma unroll
    for (int j = 0; j < 3; ++j)
      R[i][j] = V[i][0] * u0[j] + V[i][1] * u1[j] + V[i][2] * u2[j];

  // constant column of D: c = tbar - R*tbar  (folds centering + re-centroid into C matrix)
  const float cs0 = tb0 - (R[0][0]*tb0 + R[0][1]*tb1 + R[0][2]*tb2);
  const float cs1 = tb1 - (R[1][0]*tb0 + R[1][1]*tb1 + R[1][2]*tb2);
  const float cs2 = tb2 - (R[2][0]*tb0 + R[2][1]*tb1 + R[2][2]*tb2);

  // ---- apply phase: out = R * tc + c  via V_WMMA_F32_16X16X4_F32, 16 points per WMMA ----
  const int lane = tid & 31;
  const int wv   = tid >> 5;
  const int lr   = lane & 15;
  const bool hi  = lane >= 16;
  float* ob = out + (size_t)bb * (NPTS * 3);

#if __has_builtin(__builtin_amdgcn_wmma_f32_16x16x4_f32)
  // A (16x4): lanes 0-15 hold A[m][0],A[m][1]; lanes 16-31 hold A[m][2],A[m][3]
  float ar0 = hi ? R[0][2] : R[0][0];
  float ar1 = hi ? R[1][2] : R[1][0];
  float ar2 = hi ? R[2][2] : R[2][0];
  float a0 = (lr == 0) ? ar0 : (lr == 1) ? ar1 : (lr == 2) ? ar2 : 0.0f;
  float as0 = hi ? 0.0f : R[0][1];
  float as1 = hi ? 0.0f : R[1][1];
  float as2 = hi ? 0.0f : R[2][1];
  float a1 = (lr == 0) ? as0 : (lr == 1) ? as1 : (lr == 2) ? as2 : 0.0f;
  v2f Av = { a0, a1 };

  // C (16x16 f32): VGPR i, lanes 0-15 = row i; lanes 16-31 = row 8+i (unused -> 0)
  v8f Cv;
  Cv[0] = hi ? 0.0f : cs0;
  Cv[1] = hi ? 0.0f : cs1;
  Cv[2] = hi ? 0.0f : cs2;
  Cv[3] = 0.0f; Cv[4] = 0.0f; Cv[5] = 0.0f; Cv[6] = 0.0f; Cv[7] = 0.0f;

#pragma unroll
  for (int ch = 0; ch < 4; ++ch) {
    const int p = wv * 64 + ch * 16 + lr;
    // B (4x16): lane n<16 holds B[0][n],B[1][n]; lane 16+n holds B[2][n],B[3][n]
    const float bv0 = sT[3 * p + (hi ? 2 : 0)];
    const float bv1r = sT[3 * p + 1];
    const float bv1 = hi ? 0.0f : bv1r;
    v2f Bv = { bv0, bv1 };
    v8f D = __builtin_amdgcn_wmma_f32_16x16x4_f32(false, Av, false, Bv,
                                                  (short)0, Cv, false, false);
    if (!hi) {
      float* o = ob + 3 * p;
      __builtin_nontemporal_store(D[0], o + 0);
      __builtin_nontemporal_store(D[1], o + 1);
      __builtin_nontemporal_store(D[2], o + 2);
    }
  }
#else
  // VALU fallback: each thread rotates its own 2 staged points
#pragma unroll
  for (int k = 0; k < 2; ++k) {
    const int p = 2 * tid + k;
    const float x = sT[3 * p], y = sT[3 * p + 1], z = sT[3 * p + 2];
    float* o = ob + 3 * p;
    o[0] = R[0][0] * x + R[0][1] * y + R[0][2] * z + cs0;
    o[1] = R[1][0] * x + R[1][1] * y + R[1][2] * z + cs1;
    o[2] = R[2][0] * x + R[2][1] * y + R[2][2] * z + cs2;
  }
  (void)lane; (void)wv; (void)lr; (void)hi;
#endif
}

extern "C" void kernel_launch(void* const* d_in, const int* in_sizes, int n_in,
                              void* d_out, int out_size, void* d_ws, size_t ws_size,
                              hipStream_t stream) {
  const float* pred = (const float*)d_in[0];
  const float* tru  = (const float*)d_in[1];
  const float* wgt  = (const float*)d_in[2];
  const unsigned char* msk = (const unsigned char*)d_in[3];
  float* out = (float*)d_out;

  const int nb = in_sizes[0] / (NPTS * 3);   // batch count from pred_coords size
  wra_kernel<<<dim3(nb), dim3(TPB), 0, stream>>>(pred, tru, wgt, msk, out);
  (void)n_in; (void)out_size; (void)d_ws; (void)ws_size;
}